// EdgeAttrHeteroConv_13091060318486
// MI455X (gfx1250) — compile-verified
//
#include <hip/hip_runtime.h>
#include <cstddef>
#include <cstdint>

// Problem sizes (fixed by the reference)
constexpr int NC  = 20000;   // chemicals
constexpr int NG  = 30000;   // genes
constexpr int NGO = 15000;   // go terms
constexpr int NP  = 5000;    // pathways
constexpr int NE  = 200000;  // edges per type
constexpr int CD  = 256;     // channel dim

typedef float v2f __attribute__((ext_vector_type(2)));
typedef float v8f __attribute__((ext_vector_type(8)));

__device__ __forceinline__ uint32_t lds_byte_off(const void* p) {
    // addrspace(3) offset lives in the low 32 bits of the generic address
    return (uint32_t)(uintptr_t)p;
}

// ---------------------------------------------------------------------------
// Utility: zero-fill (aggregation buffers) -- graph-capture safe
// ---------------------------------------------------------------------------
__global__ void zero_f4(float4* __restrict__ p, long n4) {
    long i = (long)blockIdx.x * blockDim.x + threadIdx.x;
    long stride = (long)gridDim.x * blockDim.x;
    float4 z = make_float4(0.f, 0.f, 0.f, 0.f);
    for (; i < n4; i += stride) p[i] = z;
}

// ---------------------------------------------------------------------------
// Dense GEMM: Cout[M,256] = alpha*(A1 (+A2)) @ W[256,256] + bias
// Block = 64-row strip x 256 cols. 8 waves; each wave: 4 row-strips x 2 col
// tiles = 8 accumulators, so each B fragment is reused by 4 WMMAs.
// V_WMMA_F32_16X16X4_F32 fragments:
//   A (16x4): lane l: M=l&15, VGPR v holds K = k0 + v + 2*(l>>4)
//   B (4x16): lane l: N=l&15, VGPR v holds K = k0 + v + 2*(l>>4)
//   C/D: VGPR r -> M = r + 8*(l>>4), N = l&15
// A tile staged via gfx1250 async direct-to-LDS loads (ASYNCcnt).
// ---------------------------------------------------------------------------
template <bool HAS_A2>
__global__ __launch_bounds__(256) void gemm256_wmma(
    const float* __restrict__ A1, const float* __restrict__ A2, float alpha,
    const float* __restrict__ W, const float* __restrict__ bias,
    float* __restrict__ Cout, int M)
{
    constexpr int LDA = 260;  // 260 % 64 == 4 -> rows spread across banks
    constexpr int MT  = 64;   // rows per block
    __shared__ alignas(16) float As[MT * LDA];

    const int m0  = blockIdx.x * MT;
    const int tid = threadIdx.x;

    // ---- Stage A tile into LDS: 64 rows x 64 float4 = 4096 float4 ----
    bool used_async = false;
#pragma unroll
    for (int i = 0; i < 16; ++i) {
        int idx4 = tid + i * 256;
        int r = idx4 >> 6, c4 = idx4 & 63;
        int m = m0 + r;
        float4* dst = (float4*)&As[r * LDA + c4 * 4];
        if (!HAS_A2) {
            if (m < M) {
                const float* g = A1 + (size_t)m * CD + c4 * 4;
                asm volatile("global_load_async_to_lds_b128 %0, %1, off"
                             :: "v"(lds_byte_off(dst)), "v"(g) : "memory");
                used_async = true;
            } else {
                *dst = make_float4(0.f, 0.f, 0.f, 0.f);
            }
        } else {
            float4 v = make_float4(0.f, 0.f, 0.f, 0.f);
            if (m < M) {
                float4 va = *(const float4*)(A1 + (size_t)m * CD + c4 * 4);
                float4 vb = *(const float4*)(A2 + (size_t)m * CD + c4 * 4);
                v.x = (va.x + vb.x) * alpha;
                v.y = (va.y + vb.y) * alpha;
                v.z = (va.z + vb.z) * alpha;
                v.w = (va.w + vb.w) * alpha;
            }
            *dst = v;
        }
    }
    if (!HAS_A2 && used_async) {
        asm volatile("s_wait_asynccnt 0" ::: "memory");
    }
    __syncthreads();

    const int wave = tid >> 5, lane = tid & 31;
    const int lo = lane & 15, hi = lane >> 4;
    const int n0 = wave * 32;  // wave covers cols [n0, n0+32)

    v8f acc[4][2];
#pragma unroll
    for (int s = 0; s < 4; ++s)
#pragma unroll
        for (int t = 0; t < 2; ++t)
            acc[s][t] = (v8f){0.f, 0.f, 0.f, 0.f, 0.f, 0.f, 0.f, 0.f};

    const float* Wp = &W[(size_t)(2 * hi) * CD + n0 + lo];
#pragma unroll 2
    for (int k0 = 0; k0 < CD; k0 += 4) {
        v2f a[4], b[2];
#pragma unroll
        for (int s = 0; s < 4; ++s) {
            const float* Arow = &As[(s * 16 + lo) * LDA + 2 * hi];
            a[s].x = Arow[k0];
            a[s].y = Arow[k0 + 1];
        }
#pragma unroll
        for (int t = 0; t < 2; ++t) {
            b[t].x = Wp[(size_t)k0 * CD + t * 16];
            b[t].y = Wp[(size_t)(k0 + 1) * CD + t * 16];
        }
#pragma unroll
        for (int s = 0; s < 4; ++s)
#pragma unroll
            for (int t = 0; t < 2; ++t)
                acc[s][t] = __builtin_amdgcn_wmma_f32_16x16x4_f32(
                    false, a[s], false, b[t], (short)0, acc[s][t], false, false);
    }

#pragma unroll
    for (int t = 0; t < 2; ++t) {
        const float bn = bias[n0 + t * 16 + lo];
#pragma unroll
        for (int s = 0; s < 4; ++s) {
#pragma unroll
            for (int r = 0; r < 8; ++r) {
                int m = m0 + s * 16 + r + 8 * hi;
                if (m < M) Cout[(size_t)m * CD + n0 + t * 16 + lo] = acc[s][t][r] + bn;
            }
        }
    }
}

// ---------------------------------------------------------------------------
// Gate tables: the categorical gates take at most 50 (cg) / 3 (cp) values.
// ---------------------------------------------------------------------------
__global__ void gate_table_cg_kernel(const float* __restrict__ embT,
                                     const float* __restrict__ embS,
                                     const float* __restrict__ Wg,
                                     const float* __restrict__ bg,
                                     float* __restrict__ tab)
{
    int p = blockIdx.x;            // 0..49  => (a0, a1) = (p/5, p%5)
    int a0 = p / 5, a1 = p % 5;
    int c = threadIdx.x;
    float s = bg[c];
#pragma unroll 8
    for (int j = 0; j < 32; ++j) {
        s += embT[a0 * 32 + j] * Wg[j * CD + c];
        s += embS[a1 * 32 + j] * Wg[(32 + j) * CD + c];
    }
    tab[p * CD + c] = 1.f / (1.f + expf(-s));
}

__global__ void gate_table_cp_kernel(const float* __restrict__ embP,
                                     const float* __restrict__ Wg,
                                     const float* __restrict__ bg,
                                     float* __restrict__ tab)
{
    int p = blockIdx.x;            // 0..2
    int c = threadIdx.x;
    float s = bg[c];
#pragma unroll 8
    for (int j = 0; j < 32; ++j) s += embP[p * 32 + j] * Wg[j * CD + c];
    tab[p * CD + c] = 1.f / (1.f + expf(-s));
}

// ---------------------------------------------------------------------------
// Edge gather/scatter: one wave32 per edge; lane handles 8 channels (2xfloat4).
// softmax-over-heads then mean == * 0.25 exactly, so no attention math needed.
// MODE 0: cg (gate = tab[a0*5+a1]); MODE 1: cp (gate = tab[ea]); MODE 2: gg (no gate)
// ---------------------------------------------------------------------------
template <int MODE>
__global__ __launch_bounds__(256) void edge_kernel(
    const int* __restrict__ ei, int E, const int* __restrict__ ea,
    const float* __restrict__ gate_tab,
    const float* __restrict__ hsrc, const float* __restrict__ hdst,
    float* __restrict__ aggr)
{
    int w = (int)((blockIdx.x * (unsigned)blockDim.x + threadIdx.x) >> 5);
    int lane = threadIdx.x & 31;
    if (w >= E) return;
    int si = ei[w];
    int di = ei[E + w];
    const float4* hs4 = (const float4*)(hsrc + (size_t)si * CD);
    const float4* hd4 = (const float4*)(hdst + (size_t)di * CD);
    const float4* g4  = nullptr;
    if (MODE == 0) { int gi = ea[2 * w] * 5 + ea[2 * w + 1]; g4 = (const float4*)(gate_tab + (size_t)gi * CD); }
    if (MODE == 1) { int gi = ea[w];                          g4 = (const float4*)(gate_tab + (size_t)gi * CD); }
    float* ag = aggr + (size_t)di * CD;
#pragma unroll
    for (int j = 0; j < 2; ++j) {
        int q = lane * 2 + j;
        float4 s = hs4[q];
        float4 d = hd4[q];
        float4 g = (MODE == 2) ? make_float4(1.f, 1.f, 1.f, 1.f) : g4[q];
        unsafeAtomicAdd(&ag[q * 4 + 0], (s.x + d.x) * g.x * 0.25f);
        unsafeAtomicAdd(&ag[q * 4 + 1], (s.y + d.y) * g.y * 0.25f);
        unsafeAtomicAdd(&ag[q * 4 + 2], (s.z + d.z) * g.z * 0.25f);
        unsafeAtomicAdd(&ag[q * 4 + 3], (s.w + d.w) * g.w * 0.25f);
    }
}

// ---------------------------------------------------------------------------
// ce edges: continuous gate = sigmoid(gelu(ea@W1+b1) @ W2 + b2), fused:
//   Stage1: H[32x256] = gelu(...) into LDS (K=3 -> VALU)
//   Stage2: gate tile = H @ W2 via fp32 WMMA (2 row strips x 2 col tiles per
//           wave -> B reuse); after a barrier the sigmoid gate overwrites H
//   Stage3: gather h_src/h_dst, apply gate * 0.25, atomic scatter
// Block handles 32 edges; NE % 32 == 0.
// ---------------------------------------------------------------------------
__global__ __launch_bounds__(256) void edge_ce_kernel(
    const int* __restrict__ ei, int E, const float* __restrict__ ea,
    const float* __restrict__ W1, const float* __restrict__ b1,
    const float* __restrict__ W2, const float* __restrict__ b2,
    const float* __restrict__ hsrc, const float* __restrict__ hdst,
    float* __restrict__ aggr)
{
    constexpr int LDA = 260;
    __shared__ alignas(16) float Hs[32 * LDA];

    const int e0  = blockIdx.x * 32;
    const int tid = threadIdx.x;
    const int c   = tid;

    // Stage 1: per-edge 3->256 linear + exact GELU
    for (int r = 0; r < 32; ++r) {
        int e = e0 + r;
        float a0 = ea[(size_t)e * 3 + 0];
        float a1 = ea[(size_t)e * 3 + 1];
        float a2 = ea[(size_t)e * 3 + 2];
        float x = a0 * W1[c] + a1 * W1[CD + c] + a2 * W1[2 * CD + c] + b1[c];
        Hs[r * LDA + c] = 0.5f * x * (1.f + erff(x * 0.70710678118654752f));
    }
    __syncthreads();

    const int wave = tid >> 5, lane = tid & 31;
    const int lo = lane & 15, hi = lane >> 4;
    const int n0 = wave * 32;

    // Stage 2: 32x256 gate GEMM with fp32 WMMA
    v8f acc[2][2];
#pragma unroll
    for (int s = 0; s < 2; ++s)
#pragma unroll
        for (int t = 0; t < 2; ++t)
            acc[s][t] = (v8f){0.f, 0.f, 0.f, 0.f, 0.f, 0.f, 0.f, 0.f};

    const float* Wp = &W2[(size_t)(2 * hi) * CD + n0 + lo];
#pragma unroll 2
    for (int k0 = 0; k0 < CD; k0 += 4) {
        v2f a[2], b[2];
#pragma unroll
        for (int s = 0; s < 2; ++s) {
            const float* Arow = &Hs[(s * 16 + lo) * LDA + 2 * hi];
            a[s].x = Arow[k0];
            a[s].y = Arow[k0 + 1];
        }
#pragma unroll
        for (int t = 0; t < 2; ++t) {
            b[t].x = Wp[(size_t)k0 * CD + t * 16];
            b[t].y = Wp[(size_t)(k0 + 1) * CD + t * 16];
        }
#pragma unroll
        for (int s = 0; s < 2; ++s)
#pragma unroll
            for (int t = 0; t < 2; ++t)
                acc[s][t] = __builtin_amdgcn_wmma_f32_16x16x4_f32(
                    false, a[s], false, b[t], (short)0, acc[s][t], false, false);
    }
    __syncthreads();  // all waves done reading H before gate overwrites it

#pragma unroll
    for (int t = 0; t < 2; ++t) {
        const float bn = b2[n0 + t * 16 + lo];
#pragma unroll
        for (int s = 0; s < 2; ++s) {
#pragma unroll
            for (int r = 0; r < 8; ++r) {
                float g = 1.f / (1.f + expf(-(acc[s][t][r] + bn)));
                Hs[(s * 16 + r + 8 * hi) * LDA + n0 + t * 16 + lo] = g;
            }
        }
    }
    __syncthreads();

    // Stage 3: each wave processes 4 edges; lane covers 8 channels
#pragma unroll
    for (int t = 0; t < 4; ++t) {
        int r = wave * 4 + t;
        int e = e0 + r;
        int si = ei[e];
        int di = ei[E + e];
        const float4* hs4 = (const float4*)(hsrc + (size_t)si * CD);
        const float4* hd4 = (const float4*)(hdst + (size_t)di * CD);
        const float*  grow = &Hs[r * LDA];
        float* ag = aggr + (size_t)di * CD;
#pragma unroll
        for (int j = 0; j < 2; ++j) {
            int q = lane * 2 + j;
            float4 s = hs4[q];
            float4 d = hd4[q];
            const float4 g = *(const float4*)(grow + q * 4);
            unsafeAtomicAdd(&ag[q * 4 + 0], (s.x + d.x) * g.x * 0.25f);
            unsafeAtomicAdd(&ag[q * 4 + 1], (s.y + d.y) * g.y * 0.25f);
            unsafeAtomicAdd(&ag[q * 4 + 2], (s.z + d.z) * g.z * 0.25f);
            unsafeAtomicAdd(&ag[q * 4 + 3], (s.w + d.w) * g.w * 0.25f);
        }
    }
}

// ---------------------------------------------------------------------------
extern "C" void kernel_launch(void* const* d_in, const int* in_sizes, int n_in,
                              void* d_out, int out_size, void* d_ws, size_t ws_size,
                              hipStream_t stream)
{
    (void)in_sizes; (void)n_in; (void)out_size; (void)ws_size;

    const float* x_chem = (const float*)d_in[0];
    const float* x_gene = (const float*)d_in[1];
    const float* x_go   = (const float*)d_in[2];
    const float* x_pw   = (const float*)d_in[3];
    const int*   ei_cg  = (const int*)d_in[4];
    const int*   ei_cp  = (const int*)d_in[5];
    const int*   ei_ce  = (const int*)d_in[6];
    const int*   ei_gg  = (const int*)d_in[7];
    const int*   ea_cg  = (const int*)d_in[8];
    const int*   ea_cp  = (const int*)d_in[9];
    const float* ea_ce  = (const float*)d_in[10];
    const float* W_src_cg = (const float*)d_in[11]; const float* b_src_cg = (const float*)d_in[12];
    const float* W_dst_cg = (const float*)d_in[13]; const float* b_dst_cg = (const float*)d_in[14];
    const float* W_src_cp = (const float*)d_in[16]; const float* b_src_cp = (const float*)d_in[17];
    const float* W_dst_cp = (const float*)d_in[18]; const float* b_dst_cp = (const float*)d_in[19];
    const float* W_src_ce = (const float*)d_in[21]; const float* b_src_ce = (const float*)d_in[22];
    const float* W_dst_ce = (const float*)d_in[23]; const float* b_dst_ce = (const float*)d_in[24];
    const float* W_src_gg = (const float*)d_in[26]; const float* b_src_gg = (const float*)d_in[27];
    const float* W_dst_gg = (const float*)d_in[28]; const float* b_dst_gg = (const float*)d_in[29];
    const float* embT      = (const float*)d_in[31];
    const float* embS      = (const float*)d_in[32];
    const float* W_gate_cg = (const float*)d_in[33]; const float* b_gate_cg = (const float*)d_in[34];
    const float* embP      = (const float*)d_in[35];
    const float* W_gate_cp = (const float*)d_in[36]; const float* b_gate_cp = (const float*)d_in[37];
    const float* W_g1_ce   = (const float*)d_in[38]; const float* b_g1_ce   = (const float*)d_in[39];
    const float* W_g2_ce   = (const float*)d_in[40]; const float* b_g2_ce   = (const float*)d_in[41];
    const float* W_out_gene = (const float*)d_in[42]; const float* b_out_gene = (const float*)d_in[43];
    const float* W_out_go   = (const float*)d_in[44]; const float* b_out_go   = (const float*)d_in[45];
    const float* W_out_pw   = (const float*)d_in[46]; const float* b_out_pw   = (const float*)d_in[47];

    // Workspace layout (fp32 rows of 256)
    float* ws = (float*)d_ws;
    size_t off = 0;
    auto take = [&](size_t rows) { float* p = ws + off; off += rows * (size_t)CD; return p; };
    float* h_src_cg = take(NC);  float* h_dst_cg = take(NG);
    float* h_src_cp = take(NC);  float* h_dst_cp = take(NGO);
    float* h_src_ce = take(NC);  float* h_dst_ce = take(NP);
    float* h_src_gg = take(NG);  float* h_dst_gg = take(NG);
    float* aggr_cg  = take(NG);  float* aggr_gg  = take(NG);     // contiguous with next two
    float* aggr_cp  = take(NGO); float* aggr_ce  = take(NP);
    float* tab_cg   = take(50);  float* tab_cp   = take(3);

    float* out      = (float*)d_out;
    float* out_chem = out;
    float* out_gene = out + (size_t)NC * CD;
    float* out_go   = out_gene + (size_t)NG * CD;
    float* out_pw   = out_go + (size_t)NGO * CD;

    // 0) zero the (contiguous) aggregation buffers
    size_t aggrFloats = (size_t)(NG + NG + NGO + NP) * CD;
    zero_f4<<<2048, 256, 0, stream>>>((float4*)aggr_cg, (long)(aggrFloats / 4));

    // 0b) chemical passthrough
    hipMemcpyAsync(out_chem, x_chem, (size_t)NC * CD * sizeof(float),
                   hipMemcpyDeviceToDevice, stream);

    auto nb = [](int M) { return (M + 63) / 64; };

    // 1) node-level linear transforms (hoisted out of per-edge math)
    gemm256_wmma<false><<<nb(NC),  256, 0, stream>>>(x_chem, nullptr, 1.f, W_src_cg, b_src_cg, h_src_cg, NC);
    gemm256_wmma<false><<<nb(NG),  256, 0, stream>>>(x_gene, nullptr, 1.f, W_dst_cg, b_dst_cg, h_dst_cg, NG);
    gemm256_wmma<false><<<nb(NC),  256, 0, stream>>>(x_chem, nullptr, 1.f, W_src_cp, b_src_cp, h_src_cp, NC);
    gemm256_wmma<false><<<nb(NGO), 256, 0, stream>>>(x_go,   nullptr, 1.f, W_dst_cp, b_dst_cp, h_dst_cp, NGO);
    gemm256_wmma<false><<<nb(NC),  256, 0, stream>>>(x_chem, nullptr, 1.f, W_src_ce, b_src_ce, h_src_ce, NC);
    gemm256_wmma<false><<<nb(NP),  256, 0, stream>>>(x_pw,   nullptr, 1.f, W_dst_ce, b_dst_ce, h_dst_ce, NP);
    gemm256_wmma<false><<<nb(NG),  256, 0, stream>>>(x_gene, nullptr, 1.f, W_src_gg, b_src_gg, h_src_gg, NG);
    gemm256_wmma<false><<<nb(NG),  256, 0, stream>>>(x_gene, nullptr, 1.f, W_dst_gg, b_dst_gg, h_dst_gg, NG);

    // 2) categorical gate tables (50 rows / 3 rows)
    gate_table_cg_kernel<<<50, 256, 0, stream>>>(embT, embS, W_gate_cg, b_gate_cg, tab_cg);
    gate_table_cp_kernel<<<3, 256, 0, stream>>>(embP, W_gate_cp, b_gate_cp, tab_cp);

    // 3) edge gather / gate / scatter (attention collapses to *0.25)
    int edgeBlocks = NE / 8;  // one wave32 per edge, 8 waves/block
    edge_kernel<0><<<edgeBlocks, 256, 0, stream>>>(ei_cg, NE, ea_cg, tab_cg, h_src_cg, h_dst_cg, aggr_cg);
    edge_kernel<1><<<edgeBlocks, 256, 0, stream>>>(ei_cp, NE, ea_cp, tab_cp, h_src_cp, h_dst_cp, aggr_cp);
    edge_kernel<2><<<edgeBlocks, 256, 0, stream>>>(ei_gg, NE, nullptr, nullptr, h_src_gg, h_dst_gg, aggr_gg);
    edge_ce_kernel<<<NE / 32, 256, 0, stream>>>(ei_ce, NE, ea_ce, W_g1_ce, b_g1_ce, W_g2_ce, b_g2_ce,
                                                h_src_ce, h_dst_ce, aggr_ce);

    // 4) output projections
    gemm256_wmma<true ><<<nb(NG),  256, 0, stream>>>(aggr_cg, aggr_gg, 0.5f, W_out_gene, b_out_gene, out_gene, NG);
    gemm256_wmma<false><<<nb(NGO), 256, 0, stream>>>(aggr_cp, nullptr, 1.f,  W_out_go,   b_out_go,   out_go,   NGO);
    gemm256_wmma<false><<<nb(NP),  256, 0, stream>>>(aggr_ce, nullptr, 1.f,  W_out_pw,   b_out_pw,   out_pw,   NP);
}